// GroupResidualVectorQuantizer_16063177687197
// MI455X (gfx1250) — compile-verified
//
#include <hip/hip_runtime.h>
#include <hip/hip_bf16.h>

// ---------------------------------------------------------------------------
// Grouped Residual VQ for MI455X (gfx1250, wave32).
// fp32 WMMA (v_wmma_f32_16x16x4_f32) for the distance GEMM, async-to-LDS
// (ASYNCcnt) double buffering for the codebook stream, LDS-resident residual.
// B-fragment base offsets are made opaque (asm) so the DS combiner forms one
// base VGPR per code row-group and merges pairs into ds_load_2addr_b64.
// ---------------------------------------------------------------------------

typedef __attribute__((ext_vector_type(2))) float v2f;
typedef __attribute__((ext_vector_type(8))) float v8f;

#define LNUM   4
#define MNUM   4
#define KSH    512
#define KSP    256
#define KTOT   768
#define DDIM   256
#define NPM    16384
#define TILE_T 128          // tokens per block
#define RS     260          // padded LDS row stride (floats) -> conflict-free
#define NBLK   12           // 64-code blocks per layer (8 shared + 4 specific)
#define CBROWS 64

// float offsets inside d_out (tuple concatenated flat, all float32)
#define XQ_OFF   0ull
#define RES_OFF  16777216ull
#define IDX_OFF  33554432ull
#define DIST_OFF 33816576ull
#define LOSS_OFF 235143168ull

// LDS layout (float offsets)
#define SM_RES   0
#define SM_CB    (TILE_T * RS)            // 33280
#define CB_SZ    (CBROWS * RS)            // 16640
#define SM_XN    (SM_CB + 2 * CB_SZ)      // token ||res||^2
#define SM_EN    (SM_XN + TILE_T)         // code  ||e||^2 (current block)
#define SM_IDX   (SM_EN + CBROWS)         // per-token best index (int)
#define SM_RED   (SM_IDX + TILE_T)        // per-wave loss partials
#define SM_TOTAL (SM_RED + 8)             // 66888 floats = 267552 B (< 320 KB)

// ---- CDNA5 async global->LDS copy (ASYNCcnt tracked) ----------------------
__device__ __forceinline__ void async_b128(unsigned lds_off, unsigned goff,
                                           const float* base) {
  asm volatile("global_load_async_to_lds_b128 %0, %1, %2"
               :: "v"(lds_off), "v"(goff), "s"(base) : "memory");
}
__device__ __forceinline__ void wait_async_le16() {
  asm volatile("s_wait_asynccnt 16" ::: "memory");
}
__device__ __forceinline__ void wait_async_0() {
  asm volatile("s_wait_asynccnt 0" ::: "memory");
}
// Force a 32-bit value into a VGPR and hide its provenance from ISel so large
// LDS base constants are not re-folded into every dependent ds address.
__device__ __forceinline__ unsigned opaque_u32(unsigned v) {
  asm("" : "+v"(v));
  return v;
}

__device__ __forceinline__ const float* cb_block_ptr(const float* share,
                                                     const float* spec,
                                                     int l, int mod, int b) {
  if (b < 8) return share + ((size_t)l * KSH + (size_t)b * 64) * DDIM;
  return spec + (((size_t)l * MNUM + mod) * KSP + (size_t)(b - 8) * 64) * DDIM;
}

// Stage one 64x256 f32 codebook block into LDS buffer `buf` (async, 16 b128
// per thread; 256 threads cover 64 rows x 64 chunks).
__device__ __forceinline__ void stage_cb(const float* src, float* sm, int buf) {
  const unsigned t    = threadIdx.x;
  const unsigned rhi  = t >> 6;             // 0..3  (row sub-index)
  const unsigned colb = (t & 63u) * 16u;    // byte column within row
  const unsigned lds0 = (unsigned)(size_t)(sm + SM_CB + buf * CB_SZ);
#pragma unroll
  for (int j = 0; j < 16; ++j) {
    const unsigned row = (unsigned)j * 4u + rhi;     // 0..63
    async_b128(lds0 + row * (RS * 4u) + colb, row * 1024u + colb, src);
  }
}

__global__ void __launch_bounds__(256, 1)
grvq_kernel(const float* __restrict__ x,
            const float* __restrict__ share_emb,
            const float* __restrict__ spec_emb,
            float* __restrict__ out,
            float* __restrict__ ws) {
  extern __shared__ float sm[];
  int* smi = (int*)(sm + SM_IDX);

  const int bid    = blockIdx.x;
  const int mod    = bid >> 7;                 // modality (4 x 128 tiles)
  const int tile   = bid & 127;
  const size_t token0 = (size_t)mod * NPM + (size_t)tile * TILE_T;

  const int t    = threadIdx.x;
  const int lane = t & 31;
  const int wv   = t >> 5;
  const int hh   = lane >> 4;                  // lane half (A/B K-pair select)
  const int l16  = lane & 15;

  float* dist_out = out + DIST_OFF;
  float* idx_out  = out + IDX_OFF;

  // ---- stage x tile into LDS residual (row = t>>1, half-row per thread) ----
  {
    const int r = t >> 1, ch = t & 1;
    const float4* gs = (const float4*)(x + (token0 + r) * DDIM + ch * 128);
    float4* dst = (float4*)(sm + SM_RES + r * RS + ch * 128);
#pragma unroll
    for (int c = 0; c < 32; ++c) dst[c] = gs[c];
  }
  __syncthreads();

  // ---- token norms ||res||^2 ----
  if (t < TILE_T) sm[SM_XN + t] = 0.f;
  __syncthreads();
  {
    const int r = t >> 1, ch = t & 1;
    const float4* rp = (const float4*)(sm + SM_RES + r * RS + ch * 128);
    float s = 0.f;
#pragma unroll
    for (int c = 0; c < 32; ++c) {
      float4 v = rp[c];
      s += v.x * v.x + v.y * v.y + v.z * v.z + v.w * v.w;
    }
    atomicAdd(&sm[SM_XN + r], s);
  }
  __syncthreads();

  // prefetch first code block of layer 0 into buffer 0
  stage_cb(cb_block_ptr(share_emb, spec_emb, 0, mod, 0), sm, 0);

  for (int l = 0; l < LNUM; ++l) {
    // per-lane token norms for the 8 C-matrix rows (M = j + 8*half)
    float xnv[8];
#pragma unroll
    for (int j = 0; j < 8; ++j) xnv[j] = sm[SM_XN + wv * 16 + j + 8 * hh];

    float bestd[8];
    int   besti[8];
#pragma unroll
    for (int j = 0; j < 8; ++j) { bestd[j] = 3.4e38f; besti[j] = 0; }

    for (int b = 0; b < NBLK; ++b) {
      const int cur = b & 1;
      // double-buffered async prefetch of the next codebook block
      if (b + 1 < NBLK) {
        stage_cb(cb_block_ptr(share_emb, spec_emb, l, mod, b + 1), sm, cur ^ 1);
        wait_async_le16();
      } else if (l + 1 < LNUM) {
        stage_cb(cb_block_ptr(share_emb, spec_emb, l + 1, mod, 0), sm, cur ^ 1);
        wait_async_le16();
      } else {
        wait_async_0();
      }
      __syncthreads();                      // current buffer visible to all

      // code norms ||e||^2 for this 64-code block
      if (t < CBROWS) {
        const float4* rp = (const float4*)(sm + SM_CB + cur * CB_SZ + t * RS);
        float s = 0.f;
#pragma unroll
        for (int c = 0; c < 64; ++c) {
          float4 v = rp[c];
          s += v.x * v.x + v.y * v.y + v.z * v.z + v.w * v.w;
        }
        sm[SM_EN + t] = s;
      }
      __syncthreads();

      // ---- WMMA: 16 tokens x 64 codes, K = 256 (32 pairs of K=4) ----
      // A frag (16x4 f32): lane l16 = row M, VGPR pair = dims (4k+2*half, +1)
      // Opaque byte offsets: each base becomes sm_base + one VGPR, so the
      // per-kk +16B deltas go to DS immediate offsets and pairs merge into
      // ds_load_2addr_b64 with zero per-step address VALU.
      const unsigned bofs  = (unsigned)((SM_CB + cur * CB_SZ + l16 * RS + hh * 2) * 4);
      const unsigned aofs  = opaque_u32((unsigned)(((wv * 16 + l16) * RS + hh * 2) * 4));
      const unsigned bofs0 = opaque_u32(bofs);
      const unsigned bofs1 = opaque_u32(bofs + 16u * RS * 4u);
      const unsigned bofs2 = opaque_u32(bofs + 32u * RS * 4u);
      const unsigned bofs3 = opaque_u32(bofs + 48u * RS * 4u);
      const float* arow = (const float*)((const char*)sm + aofs);
      const float* b0   = (const float*)((const char*)sm + bofs0);
      const float* b1   = (const float*)((const char*)sm + bofs1);
      const float* b2   = (const float*)((const char*)sm + bofs2);
      const float* b3   = (const float*)((const char*)sm + bofs3);
      v8f acc[4];
#pragma unroll
      for (int nt = 0; nt < 4; ++nt) acc[nt] = (v8f){};
#pragma unroll 4
      for (int kk = 0; kk < 64; kk += 2) {
        const v2f a0  = *(const v2f*)(arow + kk * 4);
        const v2f a1  = *(const v2f*)(arow + kk * 4 + 4);
        const v2f b00 = *(const v2f*)(b0 + kk * 4);
        const v2f b01 = *(const v2f*)(b0 + kk * 4 + 4);
        const v2f b10 = *(const v2f*)(b1 + kk * 4);
        const v2f b11 = *(const v2f*)(b1 + kk * 4 + 4);
        const v2f b20 = *(const v2f*)(b2 + kk * 4);
        const v2f b21 = *(const v2f*)(b2 + kk * 4 + 4);
        const v2f b30 = *(const v2f*)(b3 + kk * 4);
        const v2f b31 = *(const v2f*)(b3 + kk * 4 + 4);
        acc[0] = __builtin_amdgcn_wmma_f32_16x16x4_f32(
            false, a0, false, b00, (short)0, acc[0], false, false);
        acc[1] = __builtin_amdgcn_wmma_f32_16x16x4_f32(
            false, a0, false, b10, (short)0, acc[1], false, false);
        acc[2] = __builtin_amdgcn_wmma_f32_16x16x4_f32(
            false, a0, false, b20, (short)0, acc[2], false, false);
        acc[3] = __builtin_amdgcn_wmma_f32_16x16x4_f32(
            false, a0, false, b30, (short)0, acc[3], false, false);
        acc[0] = __builtin_amdgcn_wmma_f32_16x16x4_f32(
            false, a1, false, b01, (short)0, acc[0], false, false);
        acc[1] = __builtin_amdgcn_wmma_f32_16x16x4_f32(
            false, a1, false, b11, (short)0, acc[1], false, false);
        acc[2] = __builtin_amdgcn_wmma_f32_16x16x4_f32(
            false, a1, false, b21, (short)0, acc[2], false, false);
        acc[3] = __builtin_amdgcn_wmma_f32_16x16x4_f32(
            false, a1, false, b31, (short)0, acc[3], false, false);
      }

      // d2 = ||x||^2 - 2*dot + ||e||^2 ; stream to global, track argmin
#pragma unroll
      for (int nt = 0; nt < 4; ++nt) {
        const float en  = sm[SM_EN + nt * 16 + l16];
        const int  code = b * 64 + nt * 16 + l16;
#pragma unroll
        for (int j = 0; j < 8; ++j) {
          const float d2 = xnv[j] - 2.0f * acc[nt][j] + en;
          const size_t tok = token0 + wv * 16 + j + 8 * hh;
          __builtin_nontemporal_store(
              d2, dist_out + ((tok * 4 + l) * (size_t)KTOT + code));
          if (d2 < bestd[j]) { bestd[j] = d2; besti[j] = code; }  // first-min
        }
      }
      __syncthreads();                      // done reading buffer `cur`
    }

    // ---- cross-lane argmin within each 16-lane half, first-index ties ----
#pragma unroll
    for (int j = 0; j < 8; ++j) {
      float d = bestd[j];
      int  bi = besti[j];
#pragma unroll
      for (int m = 8; m >= 1; m >>= 1) {
        const float od = __shfl_xor(d, m, 32);
        const int   oi = __shfl_xor(bi, m, 32);
        if (od < d || (od == d && oi < bi)) { d = od; bi = oi; }
      }
      if (l16 == 0) {
        const int tl = wv * 16 + j + 8 * hh;
        smi[tl] = bi;
        idx_out[(token0 + tl) * 4 + l] = (float)bi;
      }
    }
    __syncthreads();
    if (t < TILE_T) sm[SM_XN + t] = 0.f;
    __syncthreads();

    // ---- residual -= cb[best]; refresh ||res||^2 and loss partial ----
    float ss = 0.f;
    {
      const int r = t >> 1, ch = t & 1;
      const int ci = smi[r];
      const float* cv = (ci < KSH)
          ? share_emb + ((size_t)l * KSH + ci) * DDIM
          : spec_emb + (((size_t)l * MNUM + mod) * KSP + (ci - KSH)) * DDIM;
      const float4* ev = (const float4*)(cv + ch * 128);
      float4* rp = (float4*)(sm + SM_RES + r * RS + ch * 128);
#pragma unroll
      for (int c = 0; c < 32; ++c) {
        float4 rv = rp[c];
        const float4 e = ev[c];
        rv.x -= e.x; rv.y -= e.y; rv.z -= e.z; rv.w -= e.w;
        rp[c] = rv;
        ss += rv.x * rv.x + rv.y * rv.y + rv.z * rv.z + rv.w * rv.w;
      }
      atomicAdd(&sm[SM_XN + r], ss);        // next layer's token norms
    }
    // block-reduce loss partial (loss = 1.25 * mean(res_after^2))
#pragma unroll
    for (int m = 16; m >= 1; m >>= 1) ss += __shfl_xor(ss, m, 32);
    if (lane == 0) sm[SM_RED + wv] = ss;
    __syncthreads();
    if (t == 0) {
      float s = 0.f;
#pragma unroll
      for (int w = 0; w < 8; ++w) s += sm[SM_RED + w];
      ws[(size_t)l * 512 + bid] = s;        // deterministic partials
    }
    __syncthreads();
  }

  // ---- outputs: x_q = x - residual_final ; residual_final ----
  {
    const int r = t >> 1, ch = t & 1;
    const float4* gx = (const float4*)(x + (token0 + r) * DDIM + ch * 128);
    const float4* rp = (const float4*)(sm + SM_RES + r * RS + ch * 128);
    float4* oq = (float4*)(out + XQ_OFF + (token0 + r) * DDIM + ch * 128);
    float4* orr = (float4*)(out + RES_OFF + (token0 + r) * DDIM + ch * 128);
#pragma unroll
    for (int c = 0; c < 32; ++c) {
      const float4 xv = gx[c];
      const float4 rv = rp[c];
      float4 q;
      q.x = xv.x - rv.x; q.y = xv.y - rv.y;
      q.z = xv.z - rv.z; q.w = xv.w - rv.w;
      oq[c] = q;
      orr[c] = rv;
    }
  }
}

// Deterministic loss reduction: q_losses[L,M] from 128 block partials each.
__global__ void __launch_bounds__(128)
loss_reduce_kernel(const float* __restrict__ ws, float* __restrict__ out) {
  const int lm = blockIdx.x;                // 0..15
  const int l = lm >> 2, m = lm & 3;
  __shared__ float red[4];
  const int t = threadIdx.x;                // 128 threads
  float s = ws[(size_t)l * 512 + (size_t)m * 128 + t];
#pragma unroll
  for (int k = 16; k >= 1; k >>= 1) s += __shfl_xor(s, k, 32);
  if ((t & 31) == 0) red[t >> 5] = s;
  __syncthreads();
  if (t == 0) {
    const float tot = red[0] + red[1] + red[2] + red[3];
    out[LOSS_OFF + (size_t)(l * 4 + m)] = tot * (1.25f / (16384.f * 256.f));
  }
}

extern "C" void kernel_launch(void* const* d_in, const int* in_sizes, int n_in,
                              void* d_out, int out_size, void* d_ws,
                              size_t ws_size, hipStream_t stream) {
  const float* x     = (const float*)d_in[0];
  // d_in[1] = split_index (int64) — modality segments are fixed equal sizes.
  const float* share = (const float*)d_in[2];
  const float* spec  = (const float*)d_in[3];
  float* out = (float*)d_out;
  float* ws  = (float*)d_ws;

  const size_t smbytes = (size_t)SM_TOTAL * sizeof(float);
  hipLaunchKernelGGL(grvq_kernel, dim3(512), dim3(256), smbytes, stream,
                     x, share, spec, out, ws);
  hipLaunchKernelGGL(loss_reduce_kernel, dim3(16), dim3(128), 0, stream,
                     ws, out);
}